// TripleInteractionModel_50002009260180
// MI455X (gfx1250) — compile-verified
//
#include <hip/hip_runtime.h>
#include <hip/hip_bf16.h>

typedef __attribute__((ext_vector_type(2))) float v2f;
typedef __attribute__((ext_vector_type(8))) float v8f;

#define B_DIM  128
#define LEN_T  128
#define LEN_A  64
#define LEN_V  64

// One wave (32 lanes) handles one (b, 16-row t-tile). 8 waves per block -> one
// block covers one batch element b. Contraction over 'a' (64) is done with 16
// accumulating V_WMMA_F32_16X16X4_F32 steps, Wa replicated across N columns.
__global__ __launch_bounds__(256)
void triple_interaction_kernel(const float* __restrict__ T,
                               const float* __restrict__ A,
                               const float* __restrict__ V,
                               const float* __restrict__ Wv,
                               const float* __restrict__ bv,
                               const float* __restrict__ Wa,
                               const float* __restrict__ ba,
                               float* __restrict__ out)
{
    const int b    = blockIdx.x;
    const int wave = threadIdx.x >> 5;   // 0..7  -> t-tile index
    const int lane = threadIdx.x & 31;
    const int row  = lane & 15;          // M (t within tile) this lane owns in A-frag
    const int hi   = lane >> 4;          // 0 -> K slots {0,1}, 1 -> K slots {2,3}

    // ---- s_b = dot(V[b,:], Wv)  (64 elems, 2 per lane, then wave32 xor-reduce)
    float s = V[b * LEN_V + lane]      * Wv[lane] +
              V[b * LEN_V + 32 + lane] * Wv[32 + lane];
    #pragma unroll
    for (int off = 16; off >= 1; off >>= 1)
        s += __shfl_xor(s, off, 32);

    const float bvv = bv[0];
    const float bav = ba[0];

    // This lane's t-row scalar: T[b,t] * s_b (t fixed per lane within the tile)
    const float Ts = T[b * LEN_T + wave * 16 + row] * s;

    const float* Ab = A + b * LEN_A;

    // ---- accumulate D[m,n] = sum_a relu(Ts_m * A[a] + bv) * Wa[a] over a=0..63
    v8f c = {};
    #pragma unroll
    for (int kk = 0; kk < 16; ++kk) {
        // A-matrix f32 16x4 layout: lanes 0-15 hold M=lane, K={0,1};
        //                           lanes 16-31 hold M=lane-16, K={2,3}.
        // B-matrix uses the same physical K slots; fill both consistently.
        const int a0 = kk * 4 + hi * 2;
        const float av0 = Ab[a0];
        const float av1 = Ab[a0 + 1];
        v2f af, bf;
        af.x = fmaxf(fmaf(Ts, av0, bvv), 0.0f);   // X[t, a0]
        af.y = fmaxf(fmaf(Ts, av1, bvv), 0.0f);   // X[t, a0+1]
        bf.x = Wa[a0];                            // replicated across all N cols
        bf.y = Wa[a0 + 1];
        c = __builtin_amdgcn_wmma_f32_16x16x4_f32(
                /*neg_a=*/false, af, /*neg_b=*/false, bf,
                /*c_mod=*/(short)0, c, /*reuse_a=*/false, /*reuse_b=*/false);
    }

    // ---- D layout: VGPR r, lanes 0-15 -> (M=r, N=lane),
    // lanes 16-31 -> (M=r+8, N=lane-16). Every column N holds the same value
    // (Wa replicated), so read column N=0: lane 0 has rows 0..7, lane 16 rows 8..15.
    if ((lane & 15) == 0) {
        float* o = out + b * LEN_T + wave * 16 + hi * 8;
        #pragma unroll
        for (int r = 0; r < 8; ++r)
            o[r] = fmaxf(c[r] + bav, 0.0f);
    }
}

extern "C" void kernel_launch(void* const* d_in, const int* in_sizes, int n_in,
                              void* d_out, int out_size, void* d_ws, size_t ws_size,
                              hipStream_t stream) {
    const float* T  = (const float*)d_in[0];
    const float* A  = (const float*)d_in[1];
    const float* V  = (const float*)d_in[2];
    const float* Wv = (const float*)d_in[3];
    const float* bv = (const float*)d_in[4];
    const float* Wa = (const float*)d_in[5];
    const float* ba = (const float*)d_in[6];
    float* out = (float*)d_out;

    // 128 blocks (one per b) x 256 threads (8 waves -> 8 t-tiles of 16 rows)
    triple_interaction_kernel<<<dim3(B_DIM), dim3(256), 0, stream>>>(
        T, A, V, Wv, bv, Wa, ba, out);
}